// RotamerBuilder_67877663146011
// MI455X (gfx1250) — compile-verified
//
#include <hip/hip_runtime.h>
#include <math.h>
#include <stdint.h>

// ---------------------------------------------------------------------------
// RotamerBuilder for MI455X (gfx1250, wave32).
// Bandwidth-bound (~665 MB traffic, ~0.16 flop/byte -> ~28.5 us at 23.3 TB/s).
// Strategy:
//   * chi_embed (460 MB): 18 lanes per 72-bin row, one native v_exp_f32 per
//     bin, aligned non-temporal global_store_b128 (output >> 192 MB L2).
//   * aligned (72 MB out, 72 MB in): AoS (15,3) layout is de-strided through
//     LDS. The coords->LDS staging uses CDNA5 ASYNC copies
//     (global_load_async_to_lds_b128, ASYNCcnt) — no VGPR round trip, no
//     per-iteration s_wait_loadcnt serialization; a single s_wait_asynccnt 0
//     precedes the block barrier. Output is fully coalesced NT b128 stores.
//   * global_prefetch_b8 to stream ahead; WMMA intentionally unused (no
//     reduction dimension; matrix tiles would add data movement).
// ---------------------------------------------------------------------------

typedef float v4f __attribute__((ext_vector_type(4)));

#define NT_LOAD(p)      __builtin_nontemporal_load((p))
#define NT_STORE(v, p)  __builtin_nontemporal_store((v), (p))

#define BLOCK 256
#define ATOMS 15
#define RES_FLOATS 45                      // 15 atoms * 3
#define TILE_FLOATS (BLOCK * RES_FLOATS)   // 11520 floats = 45 KB
#define ROT_FLOATS 15                      // R(9) + mob_com(3) + fixed_com(3)

static __device__ __forceinline__ void cross3(const float a[3], const float b[3], float o[3]) {
    o[0] = a[1]*b[2] - a[2]*b[1];
    o[1] = a[2]*b[0] - a[0]*b[2];
    o[2] = a[0]*b[1] - a[1]*b[0];
}
static __device__ __forceinline__ float dot3(const float a[3], const float b[3]) {
    return a[0]*b[0] + a[1]*b[1] + a[2]*b[2];
}

// Low 32 bits of a flat LDS address are the LDS byte offset (ISA 10.2),
// which is what GLOBAL_LOAD_ASYNC_TO_LDS's
// per-lane VDST operand expects.
static __device__ __forceinline__ uint32_t lds_byte_addr(const void* p) {
    return (uint32_t)(uintptr_t)p;
}

// CDNA5 async memory->LDS copy (CDNA5 ISA), GV addressing.
// Tracked by ASYNCcnt; completion enforced with s_wait_asynccnt.
static __device__ __forceinline__ void async_copy_b128(uint32_t lds_addr, const void* gaddr) {
    asm volatile("global_load_async_to_lds_b128 %0, %1, off"
                 :: "v"(lds_addr), "v"((uint64_t)(uintptr_t)gaddr) : "memory");
}
static __device__ __forceinline__ void async_copy_b32(uint32_t lds_addr, const void* gaddr) {
    asm volatile("global_load_async_to_lds_b32 %0, %1, off"
                 :: "v"(lds_addr), "v"((uint64_t)(uintptr_t)gaddr) : "memory");
}
static __device__ __forceinline__ void wait_asynccnt0() {
    asm volatile("s_wait_asynccnt 0x0" ::: "memory");
}

// ---------------------------------------------------------------------------
// Kernel 1: Kabsch (3-point triads) + apply to 15 atoms + NeRF extension.
// One block = 256 residues. Proper rotation built as R = u0 w0^T + u1 w1^T +
// (u0 x u1) w2^T with det(W) forced +1 — algebraically identical to the
// reference's SVD + det-flip for both signs of det(C), without dividing by
// the smallest singular value.
// ---------------------------------------------------------------------------
__global__ void __launch_bounds__(BLOCK)
rotamer_geom_kernel(const float* __restrict__ fixedp,
                    const float* __restrict__ mobilep,
                    const float* __restrict__ coords,
                    const float* __restrict__ prevp,
                    const float* __restrict__ blen,
                    const float* __restrict__ bang,
                    const float* __restrict__ dihe,
                    float* __restrict__ out_aligned,   // (N,15,3)
                    float* __restrict__ out_next,      // (N,3)
                    int n)
{
    __shared__ float sCoords[TILE_FLOATS];        // 45 KB
    __shared__ float sRot[BLOCK * ROT_FLOATS];    // 15 KB

    const int tid    = threadIdx.x;
    const int rbase  = blockIdx.x * BLOCK;
    const int nres   = min(BLOCK, n - rbase);
    const int nelems = nres * RES_FLOATS;
    const int ne4    = nelems >> 2;

    // ---- Phase A1: async-copy this block's coords tile into LDS ------------
    // Block base byte offset = rbase*180 = blockIdx*46080, 16B aligned.
    // All copies issue back-to-back (ASYNCcnt), one wait at the end.
    {
        const v4f*     src4 = (const v4f*)(coords + (size_t)rbase * RES_FLOATS);
        const uint32_t ldsb = lds_byte_addr(&sCoords[0]);
        for (int e = tid; e < ne4; e += BLOCK)
            async_copy_b128(ldsb + (uint32_t)e * 16u, src4 + e);
        for (int e = (ne4 << 2) + tid; e < nelems; e += BLOCK)
            async_copy_b32(ldsb + (uint32_t)e * 4u,
                           coords + (size_t)rbase * RES_FLOATS + e);
    }

    // ---- Phase A2: per-thread Kabsch + NeRF extension ----------------------
    if (tid < nres) {
        const int i = rbase + tid;

        const float* fp = fixedp  + (size_t)i * 9;
        const float* mp = mobilep + (size_t)i * 9;
        float fx[3][3], mx[3][3];
        #pragma unroll
        for (int r = 0; r < 3; ++r)
            #pragma unroll
            for (int c = 0; c < 3; ++c) {
                fx[r][c] = fp[r*3 + c];
                mx[r][c] = mp[r*3 + c];
            }

        float fc[3], mc[3];
        #pragma unroll
        for (int c = 0; c < 3; ++c) {
            fc[c] = (fx[0][c] + fx[1][c] + fx[2][c]) * (1.0f/3.0f);
            mc[c] = (mx[0][c] + mx[1][c] + mx[2][c]) * (1.0f/3.0f);
        }

        // cross covariance C[a][b] = sum_p mob_cen[p][a] * targ_cen[p][b]
        float C[3][3];
        #pragma unroll
        for (int a = 0; a < 3; ++a)
            #pragma unroll
            for (int b = 0; b < 3; ++b) {
                float s = 0.0f;
                #pragma unroll
                for (int p = 0; p < 3; ++p)
                    s += (mx[p][a] - mc[a]) * (fx[p][b] - fc[b]);
                C[a][b] = s;
            }

        // B = C^T C (symmetric PSD)
        float Bm[3][3];
        #pragma unroll
        for (int a = 0; a < 3; ++a)
            #pragma unroll
            for (int b = 0; b < 3; ++b) {
                float s = 0.0f;
                #pragma unroll
                for (int k = 0; k < 3; ++k) s += C[k][a] * C[k][b];
                Bm[a][b] = s;
            }

        // Jacobi eigendecomposition: B = V diag(e) V^T
        float V[3][3] = {{1.f,0.f,0.f},{0.f,1.f,0.f},{0.f,0.f,1.f}};
        const int Pp[3] = {0, 0, 1};
        const int Qq[3] = {1, 2, 2};
        #pragma unroll
        for (int sweep = 0; sweep < 4; ++sweep) {
            #pragma unroll
            for (int r = 0; r < 3; ++r) {
                const int p = Pp[r], q = Qq[r], k = 3 - p - q;
                float apq = Bm[p][q];
                float app = Bm[p][p], aqq = Bm[q][q];
                float t = 0.0f;
                if (fabsf(apq) > 1e-20f) {
                    float tau = (aqq - app) / (2.0f * apq);
                    t = copysignf(1.0f, tau) / (fabsf(tau) + sqrtf(1.0f + tau*tau));
                }
                float cr = 1.0f / sqrtf(1.0f + t*t);
                float sr = t * cr;
                Bm[p][p] = app - t * apq;
                Bm[q][q] = aqq + t * apq;
                Bm[p][q] = 0.0f; Bm[q][p] = 0.0f;
                float akp = Bm[k][p], akq = Bm[k][q];
                Bm[k][p] = cr*akp - sr*akq; Bm[p][k] = Bm[k][p];
                Bm[k][q] = sr*akp + cr*akq; Bm[q][k] = Bm[k][q];
                #pragma unroll
                for (int kk = 0; kk < 3; ++kk) {
                    float vkp = V[kk][p], vkq = V[kk][q];
                    V[kk][p] = cr*vkp - sr*vkq;
                    V[kk][q] = sr*vkp + cr*vkq;
                }
            }
        }

        // sort eigenpairs descending (columns of V)
        float ev[3] = {Bm[0][0], Bm[1][1], Bm[2][2]};
        #define CSWAP(a, b)                                                    \
            do {                                                               \
                if (ev[a] < ev[b]) {                                           \
                    float t_ = ev[a]; ev[a] = ev[b]; ev[b] = t_;               \
                    float c0 = V[0][a]; V[0][a] = V[0][b]; V[0][b] = c0;       \
                    float c1 = V[1][a]; V[1][a] = V[1][b]; V[1][b] = c1;       \
                    float c2 = V[2][a]; V[2][a] = V[2][b]; V[2][b] = c2;       \
                }                                                              \
            } while (0)
        CSWAP(0, 1); CSWAP(1, 2); CSWAP(0, 1);
        #undef CSWAP

        float w0[3] = {V[0][0], V[1][0], V[2][0]};
        float w1[3] = {V[0][1], V[1][1], V[2][1]};
        float w2[3];
        cross3(w0, w1, w2);   // force det(W) = +1 (eigvec sign is free)

        // left singular vectors
        float u0[3], u1[3], u2[3];
        #pragma unroll
        for (int a = 0; a < 3; ++a) u0[a] = C[a][0]*w0[0] + C[a][1]*w0[1] + C[a][2]*w0[2];
        float in0 = 1.0f / (sqrtf(dot3(u0, u0)) + 1e-12f);
        #pragma unroll
        for (int a = 0; a < 3; ++a) u0[a] *= in0;

        #pragma unroll
        for (int a = 0; a < 3; ++a) u1[a] = C[a][0]*w1[0] + C[a][1]*w1[1] + C[a][2]*w1[2];
        float d01 = dot3(u0, u1);
        #pragma unroll
        for (int a = 0; a < 3; ++a) u1[a] -= d01 * u0[a];
        float in1 = 1.0f / (sqrtf(dot3(u1, u1)) + 1e-12f);
        #pragma unroll
        for (int a = 0; a < 3; ++a) u1[a] *= in1;

        cross3(u0, u1, u2);   // == det(U)*u2_svd; absorbs the reference det-flip

        // R = u0 w0^T + u1 w1^T + u2 w2^T, parked in LDS for phase B
        float* rp = &sRot[tid * ROT_FLOATS];
        #pragma unroll
        for (int a = 0; a < 3; ++a)
            #pragma unroll
            for (int b = 0; b < 3; ++b)
                rp[a*3 + b] = u0[a]*w0[b] + u1[a]*w1[b] + u2[a]*w2[b];
        rp[ 9] = mc[0]; rp[10] = mc[1]; rp[11] = mc[2];
        rp[12] = fc[0]; rp[13] = fc[1]; rp[14] = fc[2];

        // ---- NeRF coordinate extension (independent of coords) -------------
        const float* pr = prevp + (size_t)i * 9;
        float r0[3] = {pr[0], pr[1], pr[2]};
        float r1[3] = {pr[3], pr[4], pr[5]};
        float r2[3] = {pr[6], pr[7], pr[8]};

        float bc[3] = {r1[0]-r2[0], r1[1]-r2[1], r1[2]-r2[2]};
        float ibc = 1.0f / (sqrtf(dot3(bc, bc)) + 1e-6f);
        bc[0] *= ibc; bc[1] *= ibc; bc[2] *= ibc;

        float ab[3] = {r1[0]-r0[0], r1[1]-r0[1], r1[2]-r0[2]};
        float ba[3];
        cross3(ab, bc, ba);
        float iba = 1.0f / (sqrtf(dot3(ba, ba)) + 1e-6f);
        ba[0] *= iba; ba[1] *= iba; ba[2] *= iba;

        float m1[3];
        cross3(ba, bc, m1);

        float L  = blen[i];
        float Aa = bang[i];
        float Dd = dihe[i];
        float sA = __sinf(Aa), cA = __cosf(Aa);
        float sD = __sinf(Dd), cD = __cosf(Dd);
        float d1 = L * cA;
        float d2 = L * sA * cD;
        float d3 = -L * sA * sD;

        float* onext = out_next + (size_t)i * 3;
        NT_STORE(r2[0] + bc[0]*d1 + m1[0]*d2 + ba[0]*d3, onext + 0);
        NT_STORE(r2[1] + bc[1]*d1 + m1[1]*d2 + ba[1]*d3, onext + 1);
        NT_STORE(r2[2] + bc[2]*d1 + m1[2]*d2 + ba[2]*d3, onext + 2);
    }

    // Wait for this wave's async LDS copies, then make the tile visible to
    // every wave in the block.
    wait_asynccnt0();
    __syncthreads();

    // ---- Phase B: transform + fully coalesced NT b128 stores ---------------
    // aligned[r][atom][j] = sum_i (coords[r][atom][i] - mc[i]) * R[i][j] + fc[j]
    {
        float* oblock = out_aligned + (size_t)rbase * RES_FLOATS;
        v4f*   o4     = (v4f*)oblock;
        for (int e4 = tid; e4 < ne4; e4 += BLOCK) {
            v4f v;
            #pragma unroll
            for (int k = 0; k < 4; ++k) {
                int e     = (e4 << 2) + k;
                int r     = (unsigned)e / 45u;
                int comp  = e - r * 45;
                int j     = (unsigned)comp % 3u;
                int abase = comp - j;
                const float* cp = &sCoords[r * RES_FLOATS + abase];
                const float* rp = &sRot[r * ROT_FLOATS];
                float x0 = cp[0] - rp[ 9];
                float x1 = cp[1] - rp[10];
                float x2 = cp[2] - rp[11];
                v[k] = x0*rp[0 + j] + x1*rp[3 + j] + x2*rp[6 + j] + rp[12 + j];
            }
            NT_STORE(v, o4 + e4);
        }
        for (int e = (ne4 << 2) + tid; e < nelems; e += BLOCK) {
            int r     = (unsigned)e / 45u;
            int comp  = e - r * 45;
            int j     = (unsigned)comp % 3u;
            int abase = comp - j;
            const float* cp = &sCoords[r * RES_FLOATS + abase];
            const float* rp = &sRot[r * ROT_FLOATS];
            float x0 = cp[0] - rp[ 9];
            float x1 = cp[1] - rp[10];
            float x2 = cp[2] - rp[11];
            NT_STORE(x0*rp[0 + j] + x1*rp[3 + j] + x2*rp[6 + j] + rp[12 + j], oblock + e);
        }
    }
}

// ---------------------------------------------------------------------------
// Kernel 2: circular Gaussian RBF, chi_embed (N,4,72).
// One float4 of bins per thread; 18 lanes cover a 72-bin row -> perfectly
// coalesced 128-bit non-temporal stores (rows are 288 B, 16 B-aligned).
// exp(-d^2/12.5) = exp2(-0.1154156 * d^2) -> one native v_exp_f32 per bin.
// ---------------------------------------------------------------------------
__global__ void __launch_bounds__(BLOCK)
rotamer_rbf_kernel(const float* __restrict__ degrees,  // (N*4)
                   float* __restrict__ chi,            // (N*4*72)
                   unsigned int total4)                // N*4*18
{
    unsigned int idx = blockIdx.x * BLOCK + threadIdx.x;
    if (idx >= total4) return;

    unsigned int row = idx / 18u;            // (residue, chi) pair
    unsigned int j0  = (idx - row * 18u) * 4u;

    // Stream-ahead prefetch of the degrees input (global_prefetch_b8).
    unsigned int nrows = total4 / 18u;
    unsigned int pf = row + 8192u;
    if (pf < nrows) __builtin_prefetch(degrees + pf, 0, 0);

    float deg = degrees[row];
    const float coef = -0.11541560f;         // -1 / (2 * 2.5^2 * ln2)

    v4f v;
    #pragma unroll
    for (int k = 0; k < 4; ++k) {
        float c = -180.0f + 5.0f * (float)(j0 + (unsigned)k);
        float d = deg - c + 180.0f;
        d -= 360.0f * floorf(d * (1.0f / 360.0f));  // == jnp.mod(diff+180, 360)
        d -= 180.0f;
        v[k] = __builtin_amdgcn_exp2f(d * d * coef);
    }
    NT_STORE(v, (v4f*)(chi + (size_t)idx * 4u));
}

// ---------------------------------------------------------------------------
extern "C" void kernel_launch(void* const* d_in, const int* in_sizes, int n_in,
                              void* d_out, int out_size, void* d_ws, size_t ws_size,
                              hipStream_t stream) {
    (void)n_in; (void)out_size; (void)d_ws; (void)ws_size;

    const int N = in_sizes[0] / 9;

    const float* fixedp  = (const float*)d_in[0];
    const float* mobilep = (const float*)d_in[1];
    const float* coords  = (const float*)d_in[2];
    const float* prevp   = (const float*)d_in[3];
    const float* blen    = (const float*)d_in[4];
    const float* bang    = (const float*)d_in[5];
    const float* dihe    = (const float*)d_in[6];
    const float* degrees = (const float*)d_in[7];

    float* out         = (float*)d_out;
    float* out_aligned = out;                       // N*45
    float* out_next    = out + (size_t)N * 45;      // N*3
    float* out_chi     = out + (size_t)N * 48;      // N*4*72

    const dim3 blk(BLOCK);

    const dim3 g1((unsigned)((N + BLOCK - 1) / BLOCK));
    rotamer_geom_kernel<<<g1, blk, 0, stream>>>(fixedp, mobilep, coords, prevp,
                                                blen, bang, dihe,
                                                out_aligned, out_next, N);

    const unsigned int total4 = (unsigned int)N * 72u;   // N*4*72/4
    const dim3 g2((total4 + BLOCK - 1u) / BLOCK);
    rotamer_rbf_kernel<<<g2, blk, 0, stream>>>(degrees, out_chi, total4);
}